// Adapthisteq_65867618452277
// MI455X (gfx1250) — compile-verified
//
#include <hip/hip_runtime.h>

// ---------------------------------------------------------------------------
// Adapthisteq with K=6: torchvision equalize on 6x6 patches. P = 36 pixels,
// step = (36 - hist[last_nz]) // 255 == 0 ALWAYS  =>  identity branch taken
// for every patch, every input.  out = float32(int32(pic)).
// => pure HBM stream: ~50 MB in + ~50 MB out, floor ~4.3 us @ 23.3 TB/s.
// We stream the load side through the CDNA5 Tensor Data Mover (one DMA per
// 16 KB tile into LDS, double buffered, TENSORcnt tracked), convert with two
// VALU ops per element, and store with b128 vector stores.
// ---------------------------------------------------------------------------

typedef unsigned int       u32;
typedef unsigned long long u64;
typedef u32 u32x4 __attribute__((ext_vector_type(4)));
typedef int i32x4 __attribute__((ext_vector_type(4)));
typedef int i32x8 __attribute__((ext_vector_type(8)));

#define TILE 4096   // floats per TDM tile = 16 KB
#define TPB  256    // 8 waves (wave32)

#if __has_builtin(__builtin_amdgcn_tensor_load_to_lds)
#define HAVE_TDM 1
#else
#define HAVE_TDM 0
#endif

__device__ __forceinline__ void cvt_store4(float4* __restrict__ dst, const float4 v) {
    float4 r;
    r.x = (float)(int)v.x;   // int32 truncation then back to float, per reference
    r.y = (float)(int)v.y;
    r.z = (float)(int)v.z;
    r.w = (float)(int)v.w;
    *dst = r;
}

#if HAVE_TDM
// Build a 1-D tensor-DMA descriptor and issue TENSOR_LOAD_TO_LDS.
// D# group0: [1:0]=count=1, [63:32]=lds_addr, [120:64]=global_addr, [127:126]=type=2
// D# group1: [17:16]=data_size(2 => 4B), [79:48]=tensor_dim0, [127:112]=tile_dim0
// Groups 2/3 (and the trailing i32x8) are unused for a 1-D tile -> zeros.
__device__ __forceinline__ void tdm_load_tile(const float* gsrc, u32 lds_byte_off, u32 nelem) {
    u64 ga = (u64)(const void*)gsrc;
    u32x4 g0;
    g0.x = 1u;                                              // count=1 (valid), user mode
    g0.y = lds_byte_off;                                    // lds_addr (bytes)
    g0.z = (u32)ga;                                         // global_addr[31:0]
    g0.w = (u32)((ga >> 32) & 0x01FFFFFFu) | (2u << 30);    // global_addr[56:32] | type=2
    i32x8 g1;
    g1[0] = (int)(2u << 16);                                // data_size = 4 bytes
    g1[1] = (int)(nelem << 16);                             // tensor_dim0[15:0]  (bits 63:48)
    g1[2] = (int)(nelem >> 16);                             // tensor_dim0[31:16] (bits 79:64)
    g1[3] = (int)(nelem << 16);                             // tile_dim0          (bits 127:112)
    g1[4] = 0; g1[5] = 0; g1[6] = 0; g1[7] = 0;             // tile_dim1/2=0 (1-D), strides=0
    i32x4 z4 = {0, 0, 0, 0};
    i32x8 z8 = {0, 0, 0, 0, 0, 0, 0, 0};
    // 6-arg form (this toolchain): (g0, g1, g2, g3, extra, cpol)
    __builtin_amdgcn_tensor_load_to_lds(g0, g1, z4, z4, z8, 0);
}

__global__ void __launch_bounds__(TPB)
adapthisteq_copy_tdm(const float* __restrict__ in, float* __restrict__ out,
                     int ntiles, int n) {
    __shared__ float buf[2][TILE];                  // 32 KB of the 320 KB WGP LDS
    const int  tid   = threadIdx.x;
    const bool wave0 = (tid < 32);                  // TDM issue is per-wave; issue once
    const int  stride = gridDim.x;
    int t = blockIdx.x;

    // Prologue: kick off the first tile's DMA.
    if (t < ntiles && wave0)
        tdm_load_tile(in + (u64)t * TILE, (u32)(u64)(void*)&buf[0][0], TILE);

    int par = 0;
    for (; t < ntiles; t += stride) {
        const int tn = t + stride;
        if (wave0) {
            if (tn < ntiles) {
                // Issue next tile into the other buffer, then wait for oldest.
                tdm_load_tile(in + (u64)tn * TILE, (u32)(u64)(void*)&buf[par ^ 1][0], TILE);
                __builtin_amdgcn_s_wait_tensorcnt(1);
            } else {
                __builtin_amdgcn_s_wait_tensorcnt(0);
            }
        }
        __syncthreads();                            // tile t is now in LDS for all waves

        const float4* __restrict__ s = (const float4*)&buf[par][0];
        float4*       __restrict__ d = (float4*)(out + (u64)t * TILE);
#pragma unroll
        for (int k = 0; k < TILE / 4 / TPB; ++k)    // 4 x b128 per thread
            cvt_store4(&d[tid + k * TPB], s[tid + k * TPB]);

        __syncthreads();                            // buffer reused by DMA next iteration
        par ^= 1;
    }

    // Scalar tail (n - ntiles*TILE < TILE elements).
    for (int i = ntiles * TILE + blockIdx.x * TPB + tid; i < n; i += stride * TPB)
        out[i] = (float)(int)in[i];
}
#endif  // HAVE_TDM

// Fallback / reference path: plain b128 streaming copy with cvt.
__global__ void __launch_bounds__(TPB)
adapthisteq_copy_plain(const float* __restrict__ in, float* __restrict__ out, int n) {
    const int n4 = n >> 2;
    const float4* __restrict__ in4 = (const float4*)in;
    float4*       __restrict__ out4 = (float4*)out;
    const int gstride = gridDim.x * TPB;
    for (int i = blockIdx.x * TPB + threadIdx.x; i < n4; i += gstride) {
        float4 v = in4[i];
        cvt_store4(&out4[i], v);
    }
    for (int i = (n4 << 2) + blockIdx.x * TPB + threadIdx.x; i < n; i += gstride)
        out[i] = (float)(int)in[i];
}

extern "C" void kernel_launch(void* const* d_in, const int* in_sizes, int n_in,
                              void* d_out, int out_size, void* d_ws, size_t ws_size,
                              hipStream_t stream) {
    const float* in  = (const float*)d_in[0];
    float*       out = (float*)d_out;
    const int n = in_sizes[0];                      // 3*2046*2046 = 12,558,348
#if HAVE_TDM
    const int ntiles = n / TILE;                    // 3066 full 16 KB tiles
    int blocks = ntiles < 1024 ? (ntiles > 0 ? ntiles : 1) : 1024;
    adapthisteq_copy_tdm<<<blocks, TPB, 0, stream>>>(in, out, ntiles, n);
#else
    int blocks = ((n >> 2) + TPB - 1) / TPB;
    if (blocks > 4096) blocks = 4096;
    if (blocks < 1)    blocks = 1;
    adapthisteq_copy_plain<<<blocks, TPB, 0, stream>>>(in, out, n);
#endif
}